// SS3D_13829794693758
// MI455X (gfx1250) — compile-verified
//
#include <hip/hip_runtime.h>
#include <math.h>

// ---------------------------------------------------------------------------
// SS3D (VMamba-style 3D selective scan) for MI455X / gfx1250, fp32 end-to-end.
// GEMMs use V_WMMA_F32_16X16X4_F32 (dual accumulators for ILP); scan uses
// LDS-tiled recurrence with CDNA5 async global<->LDS copies when available.
// ---------------------------------------------------------------------------

typedef __attribute__((ext_vector_type(2))) float v2f;
typedef __attribute__((ext_vector_type(8))) float v8f;

#define L_TOT   4096      // 16*16*16
#define DN      256       // D_INNER
#define NB      2         // batch
#define KDIR    12
#define NSTATE  16
#define DTR     8
#define CROWS   40        // DT_RANK + 2*D_STATE

#if defined(__gfx1250__) && __has_builtin(__builtin_amdgcn_global_load_async_to_lds_b32) && __has_builtin(__builtin_amdgcn_s_wait_asynccnt)
#define ASYNC_LDS_LOAD 1
#endif
#if defined(__gfx1250__) && __has_builtin(__builtin_amdgcn_global_store_async_from_lds_b32) && __has_builtin(__builtin_amdgcn_s_wait_asynccnt)
#define ASYNC_LDS_STORE 1
#endif

#if defined(ASYNC_LDS_LOAD) || defined(ASYNC_LDS_STORE)
// builtins are typed on int pointers: global side addrspace(1), LDS side addrspace(3)
typedef __attribute__((address_space(1))) int gi32;
typedef __attribute__((address_space(3))) int li32;
#define AS_G32(p) ((gi32*)(void*)(p))
#define AS_L32(p) ((li32*)(void*)(p))
#endif

// local (0-based) axis orders of (D,H,W) used by the 6 scan permutations
__device__ const int ORD[6][3] = {
    {0,1,2},{0,2,1},{1,0,2},{1,2,0},{2,0,1},{2,1,0}
};

// ---------------------------------------------------------------------------
// Kernel 1: in_proj  Y(8192x512) = X(8192x128) @ W(512x128)^T
// writes xc transposed to (B,Dn,L) and z as (B*L, Dn)
// ---------------------------------------------------------------------------
__global__ void __launch_bounds__(256)
k_inproj(const float* __restrict__ X, const float* __restrict__ W,
         float* __restrict__ xc, float* __restrict__ z)
{
    const int wave = threadIdx.x >> 5;
    const int lane = threadIdx.x & 31;
    const int tile = blockIdx.x * 8 + wave;          // 16384 tiles total
    const int mt = tile >> 5;                        // 512 M-tiles
    const int nt = tile & 31;                        // 32 N-tiles
    const int m0 = mt << 4, n0 = nt << 4;

    const int row   = lane & 15;
    const int khalf = (lane >> 4) << 1;

    const float* Arow = X + (size_t)(m0 + row) * 128;
    const float* Brow = W + (size_t)(n0 + row) * 128;

    v8f c0 = {}, c1 = {};
    #pragma unroll 2
    for (int k0 = 0; k0 < 128; k0 += 8) {
        const int kk = k0 + khalf;
        v2f a0 = { Arow[kk],     Arow[kk + 1] };
        v2f b0 = { Brow[kk],     Brow[kk + 1] };
        v2f a1 = { Arow[kk + 4], Arow[kk + 5] };
        v2f b1 = { Brow[kk + 4], Brow[kk + 5] };
        c0 = __builtin_amdgcn_wmma_f32_16x16x4_f32(false, a0, false, b0,
                                                   (short)0, c0, false, false);
        c1 = __builtin_amdgcn_wmma_f32_16x16x4_f32(false, a1, false, b1,
                                                   (short)0, c1, false, false);
    }
    const v8f c = c0 + c1;

    const int mh = (lane >> 4) << 3;
    const int n  = n0 + (lane & 15);
    #pragma unroll
    for (int r = 0; r < 8; ++r) {
        const int m  = m0 + r + mh;
        const int b_ = m >> 12;          // / 4096
        const int l  = m & 4095;
        const float v = c[r];
        if (n < DN) xc[((size_t)(b_ * DN + n) << 12) + l] = v;
        else        z[(size_t)m * DN + (n - DN)] = v;
    }
}

// ---------------------------------------------------------------------------
// Kernel 2: depthwise 3x3x3 conv + bias + SiLU on (B,Dn,16,16,16)
// ---------------------------------------------------------------------------
__global__ void __launch_bounds__(256)
k_conv(const float* __restrict__ xc, const float* __restrict__ cw,
       const float* __restrict__ cb, float* __restrict__ xv)
{
    const int idx = blockIdx.x * 256 + threadIdx.x;   // 2*256*4096 threads
    const int l  = idx & 4095;
    const int ch = (idx >> 12) & 255;
    const int b  = idx >> 20;
    const int ww = l & 15, hh = (l >> 4) & 15, dd = l >> 8;

    const float* src = xc + ((size_t)(b * DN + ch) << 12);
    const float* w   = cw + ch * 27;
    float acc = cb[ch];
    #pragma unroll
    for (int kd = 0; kd < 3; ++kd) {
        const int zd = dd + kd - 1;
        if ((unsigned)zd >= 16u) continue;
        #pragma unroll
        for (int kh = 0; kh < 3; ++kh) {
            const int zh = hh + kh - 1;
            if ((unsigned)zh >= 16u) continue;
            #pragma unroll
            for (int kw = 0; kw < 3; ++kw) {
                const int zw = ww + kw - 1;
                if ((unsigned)zw >= 16u) continue;
                acc += w[(kd * 3 + kh) * 3 + kw] * src[(zd << 8) + (zh << 4) + zw];
            }
        }
    }
    acc = acc / (1.f + __expf(-acc));                 // SiLU
    xv[idx] = acc;
}

// ---------------------------------------------------------------------------
// Kernel 3: build 12 scan views  xs(B,12,Dn,L)
// grid: (16, 256, 24) ; block 256 ; l = bx*256+tid, d = by, (b,k) = bz
// ---------------------------------------------------------------------------
__global__ void __launch_bounds__(256)
k_views(const float* __restrict__ xv, float* __restrict__ xs)
{
    const int l  = blockIdx.x * 256 + threadIdx.x;
    const int d  = blockIdx.y;
    const int bk = blockIdx.z;
    const int b  = bk / KDIR;
    const int k  = bk - b * KDIR;
    const int ko = k % 6;

    const int lv = (k < 6) ? l : (4095 - l);
    const int i0 = lv >> 8, i1 = (lv >> 4) & 15, i2 = lv & 15;
    int cc[3];
    cc[ORD[ko][0]] = i0; cc[ORD[ko][1]] = i1; cc[ORD[ko][2]] = i2;
    const int lsrc = (cc[0] << 8) + (cc[1] << 4) + cc[2];

    xs[((size_t)(bk * DN + d) << 12) + l] = xv[((size_t)(b * DN + d) << 12) + lsrc];
}

// ---------------------------------------------------------------------------
// Kernel 4: x_dbl(B,12,40,L) = xpw(k)[40x256] @ xs(b,k)[256x4096]   (WMMA)
// M padded 40->48 ; tiles = 24 * 3 * 256 = 18432 ; 8 waves/block
// ---------------------------------------------------------------------------
__global__ void __launch_bounds__(256)
k_xdbl(const float* __restrict__ xpw, const float* __restrict__ xs,
       float* __restrict__ xdbl)
{
    const int wave = threadIdx.x >> 5;
    const int lane = threadIdx.x & 31;
    const int tile = blockIdx.x * 8 + wave;
    const int bk   = tile / 768;
    const int rem  = tile - bk * 768;
    const int mt   = rem % 3;
    const int nt   = rem / 3;
    const int k    = bk % KDIR;
    const int m0 = mt << 4, n0 = nt << 4;

    const float* Wk = xpw + (size_t)k * CROWS * DN;
    const float* Xs = xs + ((size_t)bk * DN << 12);

    const int row   = lane & 15;
    const int khalf = (lane >> 4) << 1;
    const int am    = m0 + row;
    const bool mok  = (am < CROWS);
    const float* Arow = Wk + (size_t)am * DN;

    v8f c0 = {}, c1 = {};
    for (int k0 = 0; k0 < DN; k0 += 8) {
        const int kk = k0 + khalf;
        v2f a0, a1;
        a0.x = mok ? Arow[kk]     : 0.f;
        a0.y = mok ? Arow[kk + 1] : 0.f;
        a1.x = mok ? Arow[kk + 4] : 0.f;
        a1.y = mok ? Arow[kk + 5] : 0.f;
        v2f b0 = { Xs[((size_t)kk << 12) + n0 + row],
                   Xs[((size_t)(kk + 1) << 12) + n0 + row] };
        v2f b1 = { Xs[((size_t)(kk + 4) << 12) + n0 + row],
                   Xs[((size_t)(kk + 5) << 12) + n0 + row] };
        c0 = __builtin_amdgcn_wmma_f32_16x16x4_f32(false, a0, false, b0,
                                                   (short)0, c0, false, false);
        c1 = __builtin_amdgcn_wmma_f32_16x16x4_f32(false, a1, false, b1,
                                                   (short)0, c1, false, false);
    }
    const v8f c = c0 + c1;

    const int mh = (lane >> 4) << 3;
    const int n  = n0 + (lane & 15);
    #pragma unroll
    for (int r = 0; r < 8; ++r) {
        const int m = m0 + r + mh;
        if (m < CROWS)
            xdbl[((size_t)(bk * CROWS + m) << 12) + n] = c[r];
    }
}

// ---------------------------------------------------------------------------
// Kernel 5: selective scan.  One block per (b,k); thread = channel d.
// LDS-tiled over chunks of 32 timesteps; y overwrites xs in place.
// Tile staging uses GLOBAL_LOAD_ASYNC_TO_LDS / STORE_ASYNC_FROM_LDS when the
// toolchain exposes them (ASYNCcnt path), else VGPR staging.
// ---------------------------------------------------------------------------
#define CH 32
__global__ void __launch_bounds__(256)
k_scan(float* __restrict__ xs, const float* __restrict__ xdbl,
       const float* __restrict__ dtw, const float* __restrict__ dtb,
       const float* __restrict__ A_logs, const float* __restrict__ Ds)
{
    const int bk = blockIdx.x;
    const int k  = bk % KDIR;
    const int d  = threadIdx.x;
    const int wave = d >> 5, lane = d & 31;

    __shared__ float xst[DN * (CH + 1)];   // 256 rows, pad 33 -> conflict-free
    __shared__ float bct[CH * CROWS];      // [l'][c] ; broadcast reads

    const int g = k * DN + d;
    float A_[NSTATE];
    #pragma unroll
    for (int n = 0; n < NSTATE; ++n) A_[n] = -__expf(A_logs[g * NSTATE + n]);
    float w_[DTR];
    #pragma unroll
    for (int r = 0; r < DTR; ++r) w_[r] = dtw[g * DTR + r];
    const float bias = dtb[g];
    const float Dv   = Ds[g];
    float h[NSTATE];
    #pragma unroll
    for (int n = 0; n < NSTATE; ++n) h[n] = 0.f;

    float* xrow = xs + ((size_t)bk * DN << 12);
    const float* drow = xdbl + ((size_t)bk * CROWS << 12);

    for (int l0 = 0; l0 < L_TOT; l0 += CH) {
        // ---- stage xs tile (256 x 32) and x_dbl tile (40 x 32) into LDS ----
#ifdef ASYNC_LDS_LOAD
        #pragma unroll
        for (int i = 0; i < 32; ++i) {
            const int row = i * 8 + wave;
            __builtin_amdgcn_global_load_async_to_lds_b32(
                AS_G32(&xrow[((size_t)row << 12) + l0 + lane]),
                AS_L32(&xst[row * (CH + 1) + lane]), 0, 0);
        }
        #pragma unroll
        for (int e = d, j = 0; j < 5; ++j, e += 256) {
            const int c = e >> 5, li = e & 31;
            __builtin_amdgcn_global_load_async_to_lds_b32(
                AS_G32(&drow[((size_t)c << 12) + l0 + li]),
                AS_L32(&bct[li * CROWS + c]), 0, 0);
        }
        __builtin_amdgcn_s_wait_asynccnt(0);
#else
        #pragma unroll
        for (int i = 0; i < 32; ++i) {
            const int row = i * 8 + wave;
            xst[row * (CH + 1) + lane] = xrow[((size_t)row << 12) + l0 + lane];
        }
        for (int e = d; e < CH * CROWS; e += 256) {
            const int c = e >> 5, li = e & 31;
            bct[li * CROWS + c] = drow[((size_t)c << 12) + l0 + li];
        }
#endif
        // prefetch next xs chunk toward L2
        if (l0 + CH < L_TOT)
            __builtin_prefetch(&xrow[((size_t)d << 12) + l0 + CH], 0, 3);
        __syncthreads();

        // ---- recurrence over the chunk (each thread owns row d) ----
        #pragma unroll 4
        for (int li = 0; li < CH; ++li) {
            const float u  = xst[d * (CH + 1) + li];
            const float* bc = &bct[li * CROWS];
            float dt = bias;
            #pragma unroll
            for (int r = 0; r < DTR; ++r) dt += w_[r] * bc[r];
            const float delta = (dt > 20.f) ? dt : log1pf(__expf(dt));
            const float du = delta * u;
            float y = Dv * u;
            #pragma unroll
            for (int n = 0; n < NSTATE; ++n) {
                const float dA = __expf(delta * A_[n]);
                h[n] = dA * h[n] + du * bc[DTR + n];
                y += h[n] * bc[DTR + NSTATE + n];
            }
            xst[d * (CH + 1) + li] = y;       // y in place of u
        }
        __syncthreads();

        // ---- write y tile back over xs ----
#ifdef ASYNC_LDS_STORE
        #pragma unroll
        for (int i = 0; i < 32; ++i) {
            const int row = i * 8 + wave;
            __builtin_amdgcn_global_store_async_from_lds_b32(
                AS_G32(&xrow[((size_t)row << 12) + l0 + lane]),
                AS_L32(&xst[row * (CH + 1) + lane]), 0, 0);
        }
        __builtin_amdgcn_s_wait_asynccnt(0);
#else
        #pragma unroll
        for (int i = 0; i < 32; ++i) {
            const int row = i * 8 + wave;
            xrow[((size_t)row << 12) + l0 + lane] = xst[row * (CH + 1) + lane];
        }
#endif
        __syncthreads();
    }
}

// ---------------------------------------------------------------------------
// Kernel 6: un-permute/un-flip + sum over 12 directions -> yT(B*L, Dn)
// grid (4096, 2) block 256 : l = bx, b = by, d = tid (coalesced store)
// ---------------------------------------------------------------------------
__global__ void __launch_bounds__(256)
k_gather(const float* __restrict__ Y, float* __restrict__ yT)
{
    const int l = blockIdx.x;
    const int b = blockIdx.y;
    const int d = threadIdx.x;
    const int cc[3] = { l >> 8, (l >> 4) & 15, l & 15 };   // dd, hh, ww

    float acc = 0.f;
    #pragma unroll
    for (int k = 0; k < KDIR; ++k) {
        const int ko = k % 6;
        const int lk = (cc[ORD[ko][0]] << 8) + (cc[ORD[ko][1]] << 4) + cc[ORD[ko][2]];
        const int lv = (k < 6) ? lk : (4095 - lk);
        acc += Y[((size_t)((b * KDIR + k) * DN + d) << 12) + lv];
    }
    yT[((size_t)(b * L_TOT + l)) * DN + d] = acc;
}

// ---------------------------------------------------------------------------
// Kernel 7: LayerNorm(256) + SiLU(z) gate -> yn(B*L, Dn)
// ---------------------------------------------------------------------------
__global__ void __launch_bounds__(256)
k_norm(const float* __restrict__ yT, const float* __restrict__ z,
       const float* __restrict__ gw, const float* __restrict__ gb,
       float* __restrict__ yn)
{
    const int m = blockIdx.x;
    const int d = threadIdx.x;
    const float v = yT[(size_t)m * DN + d];

    __shared__ float s1[256], s2[256];
    s1[d] = v; s2[d] = v * v;
    __syncthreads();
    #pragma unroll
    for (int s = 128; s > 0; s >>= 1) {
        if (d < s) { s1[d] += s1[d + s]; s2[d] += s2[d + s]; }
        __syncthreads();
    }
    const float mu  = s1[0] * (1.f / 256.f);
    const float var = s2[0] * (1.f / 256.f) - mu * mu;
    const float rs  = rsqrtf(var + 1e-5f);
    const float xv  = (v - mu) * rs * gw[d] + gb[d];
    const float zv  = z[(size_t)m * DN + d];
    yn[(size_t)m * DN + d] = xv * (zv / (1.f + __expf(-zv)));
}

// ---------------------------------------------------------------------------
// Kernel 8: out_proj  out(8192x128) = yn(8192x256) @ opw(128x256)^T  (WMMA)
// ---------------------------------------------------------------------------
__global__ void __launch_bounds__(256)
k_outproj(const float* __restrict__ yn, const float* __restrict__ opw,
          float* __restrict__ out)
{
    const int wave = threadIdx.x >> 5;
    const int lane = threadIdx.x & 31;
    const int tile = blockIdx.x * 8 + wave;   // 4096 tiles
    const int mt = tile >> 3, nt = tile & 7;
    const int m0 = mt << 4, n0 = nt << 4;

    const int row   = lane & 15;
    const int khalf = (lane >> 4) << 1;
    const float* Arow = yn  + (size_t)(m0 + row) * DN;
    const float* Brow = opw + (size_t)(n0 + row) * DN;

    v8f c0 = {}, c1 = {};
    #pragma unroll 2
    for (int k0 = 0; k0 < DN; k0 += 8) {
        const int kk = k0 + khalf;
        v2f a0 = { Arow[kk],     Arow[kk + 1] };
        v2f b0 = { Brow[kk],     Brow[kk + 1] };
        v2f a1 = { Arow[kk + 4], Arow[kk + 5] };
        v2f b1 = { Brow[kk + 4], Brow[kk + 5] };
        c0 = __builtin_amdgcn_wmma_f32_16x16x4_f32(false, a0, false, b0,
                                                   (short)0, c0, false, false);
        c1 = __builtin_amdgcn_wmma_f32_16x16x4_f32(false, a1, false, b1,
                                                   (short)0, c1, false, false);
    }
    const v8f c = c0 + c1;

    const int mh = (lane >> 4) << 3;
    const int n  = n0 + (lane & 15);
    #pragma unroll
    for (int r = 0; r < 8; ++r) {
        const int m = m0 + r + mh;
        out[(size_t)m * 128 + n] = c[r];
    }
}

// ---------------------------------------------------------------------------
extern "C" void kernel_launch(void* const* d_in, const int* in_sizes, int n_in,
                              void* d_out, int out_size, void* d_ws, size_t ws_size,
                              hipStream_t stream)
{
    const float* x    = (const float*)d_in[0];
    const float* ipw  = (const float*)d_in[1];
    const float* cw   = (const float*)d_in[2];
    const float* cb   = (const float*)d_in[3];
    const float* xpw  = (const float*)d_in[4];
    const float* dtw  = (const float*)d_in[5];
    const float* dtb  = (const float*)d_in[6];
    const float* Alog = (const float*)d_in[7];
    const float* Ds   = (const float*)d_in[8];
    const float* ong  = (const float*)d_in[9];
    const float* onb  = (const float*)d_in[10];
    const float* opw  = (const float*)d_in[11];
    float* out = (float*)d_out;

    float* ws   = (float*)d_ws;
    float* xc   = ws;                       // 2*256*4096
    float* z    = xc   + 2097152;           // 8192*256
    float* xv   = z    + 2097152;           // 2*256*4096
    float* xs   = xv   + 2097152;           // 2*12*256*4096
    float* xdbl = xs   + 25165824;          // 2*12*40*4096
    float* yT   = xdbl + 3932160;           // 8192*256
    float* yn   = yT   + 2097152;           // 8192*256

    k_inproj <<<2048, 256, 0, stream>>>(x, ipw, xc, z);
    k_conv   <<<8192, 256, 0, stream>>>(xc, cw, cb, xv);
    k_views  <<<dim3(16, 256, 24), 256, 0, stream>>>(xv, xs);
    k_xdbl   <<<2304, 256, 0, stream>>>(xpw, xs, xdbl);
    k_scan   <<<24,   256, 0, stream>>>(xs, xdbl, dtw, dtb, Alog, Ds);
    k_gather <<<dim3(4096, 2), 256, 0, stream>>>(xs, yT);
    k_norm   <<<8192, 256, 0, stream>>>(yT, z, ong, onb, yn);
    k_outproj<<<512,  256, 0, stream>>>(yn, opw, out);
}